// UltraTinyODWithPost_84825604096169
// MI455X (gfx1250) — compile-verified
//
#include <hip/hip_runtime.h>
#include <cstdint>

// ---------------- problem constants (from reference) ----------------
constexpr int B_    = 8;
constexpr int NA_   = 3;
constexpr int NC_   = 80;
constexpr int H_    = 160;
constexpr int W_    = 160;
constexpr int HW_   = H_ * W_;          // 25600
constexpr int N_    = NA_ * HW_;        // 76800 per batch
constexpr int TOPK_ = 100;

// Stage-A tiling for the TDM path: one tensor_load_to_lds moves an
// [NC_ x TILE_PX] tile (80 ch x 256 px x 4B = 80KB), double buffered.
constexpr int TILE_PX   = 256;
constexpr int TILES_PB  = 10;                          // tiles per block
constexpr int BLKS_PP_A = HW_ / (TILE_PX * TILES_PB);  // 10 blocks per plane
constexpr int CHUNK_FLT = NC_ * TILE_PX;               // floats per tile
constexpr size_t LDSA_BYTES = 2ull * CHUNK_FLT * sizeof(float); // 160 KB

// TDM builtin availability (false in the host pass -> fallback body parsed)
#if defined(__HIP_DEVICE_COMPILE__) && __has_builtin(__builtin_amdgcn_tensor_load_to_lds)
#define HAVE_TDM 1
#else
#define HAVE_TDM 0
#endif

typedef unsigned int u32x4 __attribute__((ext_vector_type(4)));
typedef int          i32x4 __attribute__((ext_vector_type(4)));
typedef int          i32x8 __attribute__((ext_vector_type(8)));

// ---------------- device helpers ----------------
__device__ __forceinline__ float sigmoidf_(float x) {
  return 1.0f / (1.0f + __expf(-x));
}
// Faithful to reference: relu(x) + log1p(exp(-|x|))
__device__ __forceinline__ float softplusf_(float x) {
  return fmaxf(x, 0.0f) + log1pf(__expf(-fabsf(x)));
}
// Monotone float->uint key (ascending float => ascending uint)
__device__ __forceinline__ uint32_t key_of_(float f) {
  uint32_t u = __float_as_uint(f);
  uint32_t mask = (uint32_t)((int32_t)u >> 31);      // all-ones if negative
  return u ^ (mask | 0x80000000u);
}
__device__ __forceinline__ float val_of_(uint32_t k) {
  uint32_t u = (k & 0x80000000u) ? (k ^ 0x80000000u) : ~k;
  return __uint_as_float(u);
}

#if HAVE_TDM
// Issue one TDM 2-D tile load: [nch rows x tile_px cols] of 4-byte elements,
// row stride = stride_px elements, into LDS at lds_addr. D# packing per
// CDNA5 ISA ch.8 (groups 0/1; groups 2/3 zero => 2-D tensor).
__device__ __forceinline__ void tdm_issue_(uint32_t lds_addr, uint64_t gaddr) {
  const uint32_t dim0   = (uint32_t)HW_;   // tensor_dim0 (elements)
  const uint32_t dim1   = (uint32_t)NC_;   // tensor_dim1
  const uint32_t tpx    = (uint32_t)TILE_PX;
  const uint32_t nch    = (uint32_t)NC_;
  const uint64_t strd   = (uint64_t)HW_;   // tensor_dim0_stride (elements)

  u32x4 g0;
  g0.x = 1u;                                          // count=1, user mode
  g0.y = lds_addr;                                    // lds_addr (bytes)
  g0.z = (uint32_t)(gaddr & 0xFFFFFFFFu);             // global_addr[31:0]
  g0.w = (uint32_t)((gaddr >> 32) & 0x01FFFFFFu)      // global_addr[56:32]
       | (2u << 30);                                  // type=2 ("image")

  i32x8 g1;
  g1[0] = (int)(2u << 16);                            // data_size=2 (4B); mask=0
  g1[1] = (int)((dim0 & 0xFFFFu) << 16);              // abar=0 | dim0[15:0]
  g1[2] = (int)((dim0 >> 16) | ((dim1 & 0xFFFFu) << 16)); // dim0[31:16]|dim1[15:0]
  g1[3] = (int)((dim1 >> 16) | (tpx << 16));          // dim1[31:16]|tile_dim0
  g1[4] = (int)(nch & 0xFFFFu);                       // tile_dim1 | tile_dim2=0
  g1[5] = (int)(uint32_t)(strd & 0xFFFFFFFFu);        // dim0_stride[31:0]
  g1[6] = (int)(uint32_t)((strd >> 32) & 0xFFFFu);    // dim0_stride[47:32]
  g1[7] = 0;                                          // dim1_stride=0 (2-D)

  i32x4 gz = {0, 0, 0, 0};
#if __clang_major__ >= 23
  i32x8 gz8 = {0, 0, 0, 0, 0, 0, 0, 0};
  __builtin_amdgcn_tensor_load_to_lds(g0, g1, gz, gz, gz8, 0);
#else
  __builtin_amdgcn_tensor_load_to_lds(g0, g1, gz, gz, 0);
#endif
}
#endif

// ---------------- Stage A: per-location channel max + score key ----------------
// Grid: 24 planes x 10 blocks; each block streams 10 tiles of
// [80 ch x 256 px] through a double-buffered 160KB LDS window using the
// Tensor Data Mover (TENSORcnt), then reduces max/argmax from LDS.
__global__ __launch_bounds__(TILE_PX) void stageA_kernel(
    const float* __restrict__ cls,   // [B, NA*NC, H, W]
    const float* __restrict__ obj,   // [B, NA, H, W]
    const float* __restrict__ qual,  // [B, NA, H, W]
    uint32_t* __restrict__ keys,     // [B*N] sortable score keys
    int32_t*  __restrict__ bestcls)  // [B*N] argmax class
{
  extern __shared__ float lbuf[];    // 2 x [NC_][TILE_PX]

  const int tid   = threadIdx.x;
  const int plane = blockIdx.x / BLKS_PP_A;   // = b*NA + a
  const int tg    = blockIdx.x % BLKS_PP_A;
  const int pix0  = tg * TILES_PB * TILE_PX;  // first pixel of this block

  const float* planep = cls + (size_t)plane * NC_ * HW_;

#if HAVE_TDM
  const uint32_t ldsbase = (uint32_t)(uintptr_t)lbuf;  // LDS byte offset
  const bool issuer = (tid < 32);                      // wave 0 only

  if (issuer)
    tdm_issue_(ldsbase, (uint64_t)(uintptr_t)(planep + pix0));

  for (int t = 0; t < TILES_PB; ++t) {
    const int cur = t & 1;
    if (issuer) {
      if (t + 1 < TILES_PB) {
        tdm_issue_(ldsbase + (uint32_t)((1 - cur) * CHUNK_FLT * 4),
                   (uint64_t)(uintptr_t)(planep + pix0 + (t + 1) * TILE_PX));
        __builtin_amdgcn_s_wait_tensorcnt(1);   // tile t has landed
      } else {
        __builtin_amdgcn_s_wait_tensorcnt(0);
      }
    }
    __syncthreads();                             // publish tile t

    const float* lb = lbuf + cur * CHUNK_FLT;
    float m  = -3.402823466e38f;
    int   mc = 0;
    #pragma unroll 8
    for (int c = 0; c < NC_; ++c) {
      float v = lb[c * TILE_PX + tid];           // conflict-free: stride 1/lane
      if (v > m) { m = v; mc = c; }
    }

    const int pix = pix0 + t * TILE_PX + tid;
    const float so = sigmoidf_(__builtin_nontemporal_load(obj  + (size_t)plane * HW_ + pix));
    const float sq = sigmoidf_(__builtin_nontemporal_load(qual + (size_t)plane * HW_ + pix));
    // sigmoid is monotone: max over sigmoid(cls) == sigmoid(max logit)
    const float score = so * sq * sigmoidf_(m);

    const size_t oi = (size_t)plane * HW_ + pix; // == b*N + a*HW + pix
    keys[oi]    = key_of_(score);
    bestcls[oi] = mc;

    __syncthreads();                             // buf[cur] free for re-issue
  }
#else
  // Fallback (also what the host pass parses): direct non-temporal stream.
  for (int t = 0; t < TILES_PB; ++t) {
    const int pix = pix0 + t * TILE_PX + tid;
    const float* gp = planep + pix;
    float m  = -3.402823466e38f;
    int   mc = 0;
    for (int c = 0; c < NC_; ++c) {
      float v = __builtin_nontemporal_load(gp + (size_t)c * HW_);
      if (v > m) { m = v; mc = c; }
    }
    const float so = sigmoidf_(__builtin_nontemporal_load(obj  + (size_t)plane * HW_ + pix));
    const float sq = sigmoidf_(__builtin_nontemporal_load(qual + (size_t)plane * HW_ + pix));
    const float score = so * sq * sigmoidf_(m);
    const size_t oi = (size_t)plane * HW_ + pix;
    keys[oi]    = key_of_(score);
    bestcls[oi] = mc;
  }
#endif
}

// ---------------- Stage B: exact per-batch top-K + box decode ----------------
// One 1024-thread block per batch. 4-pass radix select (256-bin LDS
// histograms) finds the exact 32-bit key threshold; all candidates
// (key >= T) are compacted and bitonic-sorted on (key desc, index asc),
// matching jax.lax.top_k tie-breaking; boxes decoded only for winners.
__global__ __launch_bounds__(1024) void stageB_kernel(
    const uint32_t* __restrict__ keys,
    const int32_t*  __restrict__ bestcls,
    const float*    __restrict__ box,      // [B, NA*4, H, W]
    const float*    __restrict__ anchors,  // [NA, 2]
    float*          __restrict__ out)      // [B, TOPK, 6]
{
  __shared__ uint32_t hist[256];
  __shared__ unsigned long long cand[1024];
  __shared__ uint32_t s_sel, s_need;
  __shared__ int s_cnt;

  const int tid = threadIdx.x;
  const int b   = blockIdx.x;
  const uint32_t* kb = keys + (size_t)b * N_;

  uint32_t prefix = 0, pmask = 0, need = TOPK_;
  for (int pass = 0; pass < 4; ++pass) {
    const int shift = 24 - 8 * pass;
    if (tid < 256) hist[tid] = 0u;
    __syncthreads();
    for (int i = tid; i < N_; i += 1024) {
      uint32_t k = kb[i];
      if ((k & pmask) == prefix) atomicAdd(&hist[(k >> shift) & 255u], 1u);
    }
    __syncthreads();
    if (tid == 0) {
      uint32_t cum = 0, sel = 0, nn = need;
      for (int bin = 255; bin >= 0; --bin) {
        uint32_t c = hist[bin];
        if (cum + c >= need) { sel = (uint32_t)bin; nn = need - cum; break; }
        cum += c;
      }
      s_sel = sel; s_need = nn;
    }
    __syncthreads();
    prefix |= s_sel << shift;
    pmask  |= 0xFFu << shift;
    need = s_need;
    __syncthreads();
  }
  const uint32_t T = prefix;   // exact key of the K-th element

  if (tid == 0) s_cnt = 0;
  cand[tid] = 0ull;            // padding sorts last (real keys >= 0x8000_0000)
  __syncthreads();

  for (int i = tid; i < N_; i += 1024) {
    uint32_t k = kb[i];
    if (k >= T) {
      int pos = atomicAdd(&s_cnt, 1);
      if (pos < 1024)
        cand[pos] = ((unsigned long long)k << 32) | (uint32_t)(N_ - 1 - i);
    }
  }
  __syncthreads();

  // bitonic sort, descending on combined (key, N-1-idx) => key desc, idx asc
  for (int ksz = 2; ksz <= 1024; ksz <<= 1) {
    for (int j = ksz >> 1; j > 0; j >>= 1) {
      int ixj = tid ^ j;
      if (ixj > tid) {
        unsigned long long a = cand[tid], c = cand[ixj];
        bool desc = ((tid & ksz) == 0);
        if ((a < c) == desc) { cand[tid] = c; cand[ixj] = a; }
      }
      __syncthreads();
    }
  }

  if (tid < TOPK_) {
    unsigned long long e = cand[tid];
    uint32_t k  = (uint32_t)(e >> 32);
    int idx     = N_ - 1 - (int)(uint32_t)(e & 0xFFFFFFFFu);
    float score = val_of_(k);

    int a = idx / HW_;
    int p = idx % HW_;
    int y = p / W_;
    int x = p % W_;

    int cls = bestcls[(size_t)b * N_ + idx];

    const float* bx = box + ((size_t)b * NA_ + a) * 4 * HW_ + p;
    float tx = bx[0 * (size_t)HW_];
    float ty = bx[1 * (size_t)HW_];
    float tw = bx[2 * (size_t)HW_];
    float th = bx[3 * (size_t)HW_];

    float cx = (sigmoidf_(tx) + (float)x) / (float)W_;
    float cy = (sigmoidf_(ty) + (float)y) / (float)H_;
    float bw = anchors[a * 2 + 0] * softplusf_(tw);
    float bh = anchors[a * 2 + 1] * softplusf_(th);

    float* o = out + ((size_t)b * TOPK_ + tid) * 6;
    o[0] = score;
    o[1] = (float)cls;
    o[2] = cx;
    o[3] = cy;
    o[4] = bw;
    o[5] = bh;
  }
}

// ---------------- launch ----------------
extern "C" void kernel_launch(void* const* d_in, const int* in_sizes, int n_in,
                              void* d_out, int out_size, void* d_ws, size_t ws_size,
                              hipStream_t stream) {
  (void)in_sizes; (void)n_in; (void)out_size; (void)ws_size;
  const float* box  = (const float*)d_in[0];
  const float* obj  = (const float*)d_in[1];
  const float* qual = (const float*)d_in[2];
  const float* cls  = (const float*)d_in[3];
  const float* anch = (const float*)d_in[4];

  uint32_t* keys = (uint32_t*)d_ws;                                   // B*N u32
  int32_t*  bcls = (int32_t*)((char*)d_ws +
                              sizeof(uint32_t) * (size_t)B_ * N_);    // B*N i32

  stageA_kernel<<<B_ * NA_ * BLKS_PP_A, TILE_PX, LDSA_BYTES, stream>>>(
      cls, obj, qual, keys, bcls);
  stageB_kernel<<<B_, 1024, 0, stream>>>(
      keys, bcls, box, anch, (float*)d_out);
}